// EquivariantMPBlock_51805895524941
// MI455X (gfx1250) — compile-verified
//
#include <hip/hip_runtime.h>
#include <math.h>

typedef __attribute__((ext_vector_type(2))) float v2f;
typedef __attribute__((ext_vector_type(8))) float v8f;

#define INV_SM  0.17677669529663687f   // 1/sqrt(32)
#define C_DOT   0.10206207261596575f   // 1/sqrt(96)
#define C_SCL   0.07216878364870323f   // 1/sqrt(192)
#define C_CRS   0.05103103630798288f   // 1/sqrt(384)
#define SQRT3   1.7320508075688772f

// ---- 16x16 f32 WMMA tile: acc += A(16x32) @ B(32x16), K chained in steps of 4.
// A: element [m][k] = A[m*aRow + k*aK]   (row = node, k = feature)
// B: element [k][n] = B[k*bRow + n]
// ISA layout: A lanes 0-15 -> M=0..15 holding K={k0,k0+1}; lanes 16-31 hold K={k0+2,k0+3}.
//             B mirrors with N on lanes. C/D: VGPR r, lane L -> [r + 8*(L>=16)][L&15].
__device__ __forceinline__ v8f wmma_tile_f32(const float* __restrict__ A, int aRow, int aK,
                                             const float* __restrict__ B, int bRow,
                                             v8f acc, int lane)
{
  const int half = lane >> 4;
  const int l16  = lane & 15;
  #pragma unroll
  for (int k0 = 0; k0 < 32; k0 += 4) {
    const int ka = k0 + 2 * half;
    v2f a, b;
    a.x = A[l16 * aRow + ka * aK];
    a.y = A[l16 * aRow + (ka + 1) * aK];
    b.x = B[ka * bRow + l16];
    b.y = B[(ka + 1) * bRow + l16];
    acc = __builtin_amdgcn_wmma_f32_16x16x4_f32(false, a, false, b, (short)0, acc,
                                                false, false);
  }
  return acc;
}

// ---------------------------------------------------------------- zero scratch
__global__ void zero_kernel(float* __restrict__ p, size_t n) {
  size_t i = (size_t)blockIdx.x * blockDim.x + threadIdx.x;
  if (i < n) p[i] = 0.0f;
}

// ------------------------------------------------------- fuse tiny 32x32 weights
// Wds = W_tp_d @ W_msg_s * (C_DOT*INV_SM)
// Wsv = W_tp_s @ W_msg_v * (C_SCL*INV_SM)
// Wcv = W_tp_c @ W_msg_v * (C_CRS*INV_SM)
__global__ void fuse_w_kernel(const float* __restrict__ Wtd, const float* __restrict__ Wts,
                              const float* __restrict__ Wtc, const float* __restrict__ Wms,
                              const float* __restrict__ Wmv,
                              float* __restrict__ Wds, float* __restrict__ Wsv,
                              float* __restrict__ Wcv) {
  const int a = threadIdx.x >> 5;   // 0..31
  const int k = threadIdx.x & 31;   // 0..31
  float s1 = 0.f, s2 = 0.f, s3 = 0.f;
  #pragma unroll 8
  for (int b = 0; b < 32; b++) {
    s1 += Wtd[a * 32 + b] * Wms[b * 32 + k];
    s2 += Wts[a * 32 + b] * Wmv[b * 32 + k];
    s3 += Wtc[a * 32 + b] * Wmv[b * 32 + k];
  }
  Wds[a * 32 + k] = s1 * (C_DOT * INV_SM);
  Wsv[a * 32 + k] = s2 * (C_SCL * INV_SM);
  Wcv[a * 32 + k] = s3 * (C_CRS * INV_SM);
}

// ------------------------------------------------- stage 1: node transform (WMMA)
// pre[node*128 + 0..31]           = relu(s @ W_node_s[:, :32] * INV_SM)
// pre[node*128 + 32 + j*3 + d]    = (v_d @ W_node_v)[j]*INV_SM * sigmoid(h_s[32+j])
__global__ void node_pre_kernel(const float* __restrict__ x,
                                const float* __restrict__ Wns,   // 32x64
                                const float* __restrict__ Wnv,   // 32x32
                                float* __restrict__ pre, int ntiles) {
  const int wave = (blockIdx.x * blockDim.x + threadIdx.x) >> 5;
  const int lane = threadIdx.x & 31;
  if (wave >= ntiles) return;                     // wave-uniform: EXEC stays full
  const int half = lane >> 4, l16 = lane & 15;
  const int base = wave * 16;
  const float* As = x + (size_t)base * 128;       // s rows: stride 128, k-stride 1

  v8f z = {};
  v8f hs0 = wmma_tile_f32(As, 128, 1, Wns +  0, 64, z, lane);
  v8f hs1 = wmma_tile_f32(As, 128, 1, Wns + 16, 64, z, lane);
  v8f hs2 = wmma_tile_f32(As, 128, 1, Wns + 32, 64, z, lane);
  v8f hs3 = wmma_tile_f32(As, 128, 1, Wns + 48, 64, z, lane);

  v8f g0, g1;
  #pragma unroll
  for (int r = 0; r < 8; r++) {
    const int node = base + r + 8 * half;
    float a0 = hs0[r] * INV_SM; a0 = a0 > 0.f ? a0 : 0.f;
    float a1 = hs1[r] * INV_SM; a1 = a1 > 0.f ? a1 : 0.f;
    pre[(size_t)node * 128 + l16]      = a0;
    pre[(size_t)node * 128 + 16 + l16] = a1;
    g0[r] = 1.f / (1.f + expf(-(hs2[r] * INV_SM)));
    g1[r] = 1.f / (1.f + expf(-(hs3[r] * INV_SM)));
  }

  #pragma unroll
  for (int d = 0; d < 3; d++) {
    const float* Av = x + (size_t)base * 128 + 32 + d;   // v_d: stride 128, k-stride 3
    v8f h0 = wmma_tile_f32(Av, 128, 3, Wnv +  0, 32, z, lane);
    v8f h1 = wmma_tile_f32(Av, 128, 3, Wnv + 16, 32, z, lane);
    #pragma unroll
    for (int r = 0; r < 8; r++) {
      const int node = base + r + 8 * half;
      pre[(size_t)node * 128 + 32 + (size_t)l16 * 3 + d]        = h0[r] * INV_SM * g0[r];
      pre[(size_t)node * 128 + 32 + (size_t)(16 + l16) * 3 + d] = h1[r] * INV_SM * g1[r];
    }
  }
}

// --------------------------------------------- stage 2: edges (gather + scatter)
__global__ void edge_kernel(const float* __restrict__ pre, const float* __restrict__ pos,
                            const int* __restrict__ ei,
                            float* __restrict__ aggD, float* __restrict__ aggS,
                            float* __restrict__ aggC, float* __restrict__ deg, int E) {
  const int e = blockIdx.x * blockDim.x + threadIdx.x;
  if (e >= E) return;
  const int r = ei[e];
  const int c = ei[E + e];
  const float rx = pos[(size_t)c * 3 + 0] - pos[(size_t)r * 3 + 0];
  const float ry = pos[(size_t)c * 3 + 1] - pos[(size_t)r * 3 + 1];
  const float rz = pos[(size_t)c * 3 + 2] - pos[(size_t)r * 3 + 2];
  const float inv = rsqrtf(rx * rx + ry * ry + rz * rz + 1e-12f);
  // u = sqrt(3) * unit[(1,2,0)]
  const float ux = SQRT3 * ry * inv;
  const float uy = SQRT3 * rz * inv;
  const float uz = SQRT3 * rx * inv;
  atomicAdd(&deg[r], 1.0f);
  const float* nc = pre + (size_t)c * 128;
  float* aD = aggD + (size_t)r * 32;
  float* aS = aggS + (size_t)r * 96;
  float* aC = aggC + (size_t)r * 96;
  #pragma unroll 4
  for (int i = 0; i < 32; i++) {
    const float sj = nc[i];
    const float vx = nc[32 + i * 3 + 0];
    const float vy = nc[32 + i * 3 + 1];
    const float vz = nc[32 + i * 3 + 2];
    atomicAdd(&aD[i], vx * ux + vy * uy + vz * uz);
    atomicAdd(&aS[i * 3 + 0], sj * ux);
    atomicAdd(&aS[i * 3 + 1], sj * uy);
    atomicAdd(&aS[i * 3 + 2], sj * uz);
    atomicAdd(&aC[i * 3 + 0], vy * uz - vz * uy);
    atomicAdd(&aC[i * 3 + 1], vz * ux - vx * uz);
    atomicAdd(&aC[i * 3 + 2], vx * uy - vy * ux);
  }
}

// ------------------------------------- stage 3a: post-aggregation matmuls (WMMA)
__global__ void node_post_kernel(const float* __restrict__ aggD, const float* __restrict__ aggS,
                                 const float* __restrict__ aggC, const float* __restrict__ deg,
                                 const float* __restrict__ Wds, const float* __restrict__ Wsv,
                                 const float* __restrict__ Wcv,
                                 float* __restrict__ out, int ntiles) {
  const int wave = (blockIdx.x * blockDim.x + threadIdx.x) >> 5;
  const int lane = threadIdx.x & 31;
  if (wave >= ntiles) return;
  const int half = lane >> 4, l16 = lane & 15;
  const int base = wave * 16;

  float dinv[8];
  #pragma unroll
  for (int r = 0; r < 8; r++)
    dinv[r] = 1.0f / fmaxf(deg[base + r + 8 * half], 1.0f);

  v8f z = {};
  const float* Ad = aggD + (size_t)base * 32;
  v8f s0 = wmma_tile_f32(Ad, 32, 1, Wds +  0, 32, z, lane);
  v8f s1 = wmma_tile_f32(Ad, 32, 1, Wds + 16, 32, z, lane);
  #pragma unroll
  for (int r = 0; r < 8; r++) {
    const int node = base + r + 8 * half;
    out[(size_t)node * 128 + l16]      = s0[r] * dinv[r];
    out[(size_t)node * 128 + 16 + l16] = s1[r] * dinv[r];
  }

  #pragma unroll
  for (int d = 0; d < 3; d++) {
    const float* As = aggS + (size_t)base * 96 + d;   // stride 96, k-stride 3
    const float* Ac = aggC + (size_t)base * 96 + d;
    v8f a0 = {}, a1 = {};
    a0 = wmma_tile_f32(As, 96, 3, Wsv +  0, 32, a0, lane);
    a0 = wmma_tile_f32(Ac, 96, 3, Wcv +  0, 32, a0, lane);
    a1 = wmma_tile_f32(As, 96, 3, Wsv + 16, 32, a1, lane);
    a1 = wmma_tile_f32(Ac, 96, 3, Wcv + 16, 32, a1, lane);
    #pragma unroll
    for (int r = 0; r < 8; r++) {
      const int node = base + r + 8 * half;
      out[(size_t)node * 128 + 32 + (size_t)l16 * 3 + d]        = a0[r] * dinv[r];
      out[(size_t)node * 128 + 32 + (size_t)(16 + l16) * 3 + d] = a1[r] * dinv[r];
    }
  }
}

// -------------------------------- stage 3b: LayerNorm + residual (wave per node)
__global__ void ln_kernel(const float* __restrict__ x, const float* __restrict__ gamma,
                          const float* __restrict__ beta, float* __restrict__ out, int N) {
  const int node = (blockIdx.x * blockDim.x + threadIdx.x) >> 5;
  const int lane = threadIdx.x & 31;
  if (node >= N) return;
  const float4 v = ((const float4*)(out + (size_t)node * 128))[lane];
  float s  = v.x + v.y + v.z + v.w;
  float sq = v.x * v.x + v.y * v.y + v.z * v.z + v.w * v.w;
  #pragma unroll
  for (int off = 16; off > 0; off >>= 1) {
    s  += __shfl_xor(s,  off, 32);
    sq += __shfl_xor(sq, off, 32);
  }
  const float mu   = s * (1.0f / 128.0f);
  const float var  = sq * (1.0f / 128.0f) - mu * mu;
  const float rstd = rsqrtf(var + 1e-5f);
  const float4 g  = ((const float4*)gamma)[lane];
  const float4 b  = ((const float4*)beta)[lane];
  const float4 xr = ((const float4*)(x + (size_t)node * 128))[lane];
  float4 o;
  o.x = (v.x - mu) * rstd * g.x + b.x + xr.x;
  o.y = (v.y - mu) * rstd * g.y + b.y + xr.y;
  o.z = (v.z - mu) * rstd * g.z + b.z + xr.z;
  o.w = (v.w - mu) * rstd * g.w + b.w + xr.w;
  ((float4*)(out + (size_t)node * 128))[lane] = o;
}

// ------------------------------------------------------------------------ launch
extern "C" void kernel_launch(void* const* d_in, const int* in_sizes, int n_in,
                              void* d_out, int out_size, void* d_ws, size_t ws_size,
                              hipStream_t stream) {
  const float* x     = (const float*)d_in[0];
  const float* pos   = (const float*)d_in[1];
  const int*   ei    = (const int*)d_in[2];
  const float* Wns   = (const float*)d_in[3];
  const float* Wnv   = (const float*)d_in[4];
  const float* Wtd   = (const float*)d_in[5];
  const float* Wts   = (const float*)d_in[6];
  const float* Wtc   = (const float*)d_in[7];
  const float* Wms   = (const float*)d_in[8];
  const float* Wmv   = (const float*)d_in[9];
  const float* gamma = (const float*)d_in[10];
  const float* beta  = (const float*)d_in[11];
  float* out = (float*)d_out;

  const int N = in_sizes[0] / 128;   // 100000 (divisible by 16)
  const int E = in_sizes[2] / 2;     // 1600000

  // workspace layout (floats): aggD[N*32] aggS[N*96] aggC[N*96] deg[N] Wds/Wsv/Wcv[1024 each]
  float* w    = (float*)d_ws;
  float* aggD = w;
  float* aggS = aggD + (size_t)N * 32;
  float* aggC = aggS + (size_t)N * 96;
  float* deg  = aggC + (size_t)N * 96;
  float* Wds  = deg + N;
  float* Wsv  = Wds + 1024;
  float* Wcv  = Wsv + 1024;

  const size_t zcount = (size_t)N * 225;   // aggD+aggS+aggC+deg
  zero_kernel<<<(int)((zcount + 1023) / 1024), 1024, 0, stream>>>(w, zcount);
  fuse_w_kernel<<<1, 1024, 0, stream>>>(Wtd, Wts, Wtc, Wms, Wmv, Wds, Wsv, Wcv);

  const int ntiles = N / 16;
  node_pre_kernel<<<(ntiles + 7) / 8, 256, 0, stream>>>(x, Wns, Wnv, out, ntiles);
  edge_kernel<<<(E + 255) / 256, 256, 0, stream>>>(out, pos, ei, aggD, aggS, aggC, deg, E);
  node_post_kernel<<<(ntiles + 7) / 8, 256, 0, stream>>>(aggD, aggS, aggC, deg,
                                                         Wds, Wsv, Wcv, out, ntiles);
  ln_kernel<<<(N + 7) / 8, 256, 0, stream>>>(x, gamma, beta, out, N);
}